// GCN_14405320311543
// MI455X (gfx1250) — compile-verified
//
#include <hip/hip_runtime.h>
#include <math.h>

#define NN 100000
#define HID 64
#define NCLS 40

typedef float v2f __attribute__((ext_vector_type(2)));
typedef float v8f __attribute__((ext_vector_type(8)));

// ---------------- degree / dinv ----------------
__global__ void init_deg_kernel(float* __restrict__ deg, int n) {
    int i = blockIdx.x * blockDim.x + threadIdx.x;
    if (i < n) deg[i] = 1.0f;   // self loop contributes 1
}

__global__ void count_deg_kernel(const long long* __restrict__ dst,
                                 float* __restrict__ deg, long long nE) {
    long long e = (long long)blockIdx.x * blockDim.x + threadIdx.x;
    if (e < nE) atomicAdd(&deg[(int)dst[e]], 1.0f);
}

__global__ void rsqrt_kernel(float* __restrict__ deg, int n) {
    int i = blockIdx.x * blockDim.x + threadIdx.x;
    if (i < n) deg[i] = rsqrtf(deg[i]);
}

__global__ void zero_kernel(float* __restrict__ p, long long n) {
    long long i = (long long)blockIdx.x * blockDim.x + threadIdx.x;
    if (i < n) p[i] = 0.0f;
}

// ---------------- fp32 WMMA GEMM: C[M,N] = A[M,K] @ B[K,N], row-major ----------------
// One wave computes one 16x16 output tile using V_WMMA_F32_16X16X4_F32.
// Two independent accumulators break the WMMA RAW chain so consecutive
// matrix ops can pipeline instead of serializing on v[acc].
// A 16x4 layout : lane&15 = M row; lane>>4 selects K pair (VGPR0=K even, VGPR1=K odd).
// B 4x16 layout : symmetric; lane&15 = N col.
// C/D           : 8 VGPRs, lane&15 = N col, (lane>>4)*8 + vgpr = M row.
__global__ void gemm_wmma_f32_kernel(const float* __restrict__ A,
                                     const float* __restrict__ B,
                                     float* __restrict__ C,
                                     int M, int K, int N) {
    int wave = blockIdx.x * (blockDim.x >> 5) + (threadIdx.x >> 5);
    int tiles_n = N >> 4;
    int tiles_m = M >> 4;
    if (wave >= tiles_m * tiles_n) return;   // whole-wave exit; EXEC uniform
    int tm = wave / tiles_n;
    int tn = wave - tm * tiles_n;
    int m0 = tm << 4, n0 = tn << 4;

    int lane  = threadIdx.x & 31;
    int idx16 = lane & 15;          // row for A, col for B/C
    int g     = lane >> 4;          // K-pair selector

    v8f acc0 = {};
    v8f acc1 = {};
    const float* arow = A + (long long)(m0 + idx16) * K + 2 * g;       // A[row][2g + k]
    const float* bcol = B + (long long)(2 * g) * N + (n0 + idx16);     // B[2g + k][col]

    #pragma unroll 4
    for (int k = 0; k < K; k += 8) {
        // prefetch the A stream one 256B step ahead (global_prefetch_b8)
        __builtin_prefetch(arow + k + 64, 0, 1);

        v2f a0 = *(const v2f*)(arow + k);                    // K block [k, k+4)
        v2f a1 = *(const v2f*)(arow + k + 4);                // K block [k+4, k+8)
        v2f b0, b1;
        b0.x = bcol[(long long)k * N];
        b0.y = bcol[(long long)k * N + N];
        b1.x = bcol[(long long)(k + 4) * N];
        b1.y = bcol[(long long)(k + 4) * N + N];
        acc0 = __builtin_amdgcn_wmma_f32_16x16x4_f32(
            false, a0, false, b0, (short)0, acc0, false, false);
        acc1 = __builtin_amdgcn_wmma_f32_16x16x4_f32(
            false, a1, false, b1, (short)0, acc1, false, false);
    }

    float* crow = C + (long long)(m0 + g * 8) * N + (n0 + idx16);
    #pragma unroll
    for (int j = 0; j < 8; ++j) crow[(long long)j * N] = acc0[j] + acc1[j];
}

// ---------------- edge scatter: agg[d] += h[s] * dinv[s]*dinv[d] ----------------
// 16 threads per edge, each handles 4 contiguous floats (float4 gather + 4 atomics).
__global__ void scatter_kernel(const float* __restrict__ h,
                               const long long* __restrict__ src,
                               const long long* __restrict__ dst,
                               const float* __restrict__ dinv,
                               float* __restrict__ agg, long long nE) {
    long long t = (long long)blockIdx.x * blockDim.x + threadIdx.x;
    long long e = t >> 4;
    if (e >= nE) return;
    int j4 = ((int)t & 15) << 2;
    int s = (int)src[e];
    int d = (int)dst[e];
    float norm = dinv[s] * dinv[d];
    const float4 hv = *(const float4*)(h + (long long)s * HID + j4);
    float* dp = agg + (long long)d * HID + j4;
    atomicAdd(dp + 0, hv.x * norm);
    atomicAdd(dp + 1, hv.y * norm);
    atomicAdd(dp + 2, hv.z * norm);
    atomicAdd(dp + 3, hv.w * norm);
}

// ---------------- self-loop + bias (+ReLU) epilogue ----------------
__global__ void epilogue_kernel(float* __restrict__ agg,
                                const float* __restrict__ h,
                                const float* __restrict__ dinv,
                                const float* __restrict__ bias,
                                long long total, int do_relu) {
    long long t = (long long)blockIdx.x * blockDim.x + threadIdx.x;
    if (t >= total) return;
    int i = (int)(t >> 6);
    int j = (int)(t & (HID - 1));
    float di = dinv[i];
    float v = agg[t] + h[t] * di * di + bias[j];
    if (do_relu) v = fmaxf(v, 0.0f);
    agg[t] = v;
}

// ---------------- fused output GEMM + softmax: one warp per node ----------------
__global__ void out_softmax_kernel(const float* __restrict__ h,
                                   const float* __restrict__ Wout,
                                   const float* __restrict__ bout,
                                   float* __restrict__ out, int n) {
    int node = blockIdx.x * (blockDim.x >> 5) + (threadIdx.x >> 5);
    if (node >= n) return;
    int lane = threadIdx.x & 31;
    int c0 = lane;            // always < 40
    int c1 = lane + 32;       // valid when lane < 8
    bool v1ok = (c1 < NCLS);

    const float* hr = h + (long long)node * HID;
    float v0 = bout[c0];
    float v1 = v1ok ? bout[c1] : -INFINITY;
    for (int k = 0; k < HID; ++k) {
        float hk = hr[k];
        v0 += hk * Wout[k * NCLS + c0];
        if (v1ok) v1 += hk * Wout[k * NCLS + c1];
    }

    float m = fmaxf(v0, v1ok ? v1 : -INFINITY);
    for (int off = 16; off > 0; off >>= 1)
        m = fmaxf(m, __shfl_xor(m, off, 32));

    float e0 = __expf(v0 - m);
    float e1 = v1ok ? __expf(v1 - m) : 0.0f;
    float s = e0 + e1;
    for (int off = 16; off > 0; off >>= 1)
        s += __shfl_xor(s, off, 32);
    float inv = 1.0f / s;

    float* orow = out + (long long)node * NCLS;
    orow[c0] = e0 * inv;
    if (v1ok) orow[c1] = e1 * inv;
}

// ---------------- driver ----------------
extern "C" void kernel_launch(void* const* d_in, const int* in_sizes, int n_in,
                              void* d_out, int out_size, void* d_ws, size_t ws_size,
                              hipStream_t stream) {
    const float*     x    = (const float*)d_in[0];
    const long long* ei   = (const long long*)d_in[1];
    const float*     W1   = (const float*)d_in[2];
    const float*     b1   = (const float*)d_in[3];
    const float*     W2   = (const float*)d_in[4];
    const float*     b2   = (const float*)d_in[5];
    const float*     Wout = (const float*)d_in[6];
    const float*     bout = (const float*)d_in[7];
    float*           out  = (float*)d_out;

    const long long E = (long long)(in_sizes[1] / 2);
    const long long* src = ei;
    const long long* dst = ei + E;

    float* dinv = (float*)d_ws;                       // NN floats (deg, then dinv)
    float* bufH = dinv + NN;                          // NN*64 floats (pre-agg features)
    float* bufA = bufH + (size_t)NN * HID;            // NN*64 floats (aggregated)

    const long long featTot = (long long)NN * HID;
    const int B256 = 256;

    // degrees -> dinv
    init_deg_kernel<<<(NN + B256 - 1) / B256, B256, 0, stream>>>(dinv, NN);
    count_deg_kernel<<<(int)((E + B256 - 1) / B256), B256, 0, stream>>>(dst, dinv, E);
    rsqrt_kernel<<<(NN + B256 - 1) / B256, B256, 0, stream>>>(dinv, NN);

    // ---- layer 1: h1 = x @ W1 ; agg ; +self +b1 ; relu ----
    {
        int tiles = (NN / 16) * (HID / 16);           // 6250 * 4 = 25000 waves
        int blocks = (tiles + 7) / 8;                 // 8 waves / block
        gemm_wmma_f32_kernel<<<blocks, 256, 0, stream>>>(x, W1, bufH, NN, 256, HID);
    }
    zero_kernel<<<(int)((featTot + B256 - 1) / B256), B256, 0, stream>>>(bufA, featTot);
    {
        long long t = E * 16;
        scatter_kernel<<<(int)((t + B256 - 1) / B256), B256, 0, stream>>>(bufH, src, dst, dinv, bufA, E);
    }
    epilogue_kernel<<<(int)((featTot + B256 - 1) / B256), B256, 0, stream>>>(bufA, bufH, dinv, b1, featTot, 1);

    // ---- layer 2: h2 = relu1 @ W2 ; agg ; +self +b2 ----
    {
        int tiles = (NN / 16) * (HID / 16);
        int blocks = (tiles + 7) / 8;
        gemm_wmma_f32_kernel<<<blocks, 256, 0, stream>>>(bufA, W2, bufH, NN, HID, HID);
    }
    zero_kernel<<<(int)((featTot + B256 - 1) / B256), B256, 0, stream>>>(bufA, featTot);
    {
        long long t = E * 16;
        scatter_kernel<<<(int)((t + B256 - 1) / B256), B256, 0, stream>>>(bufH, src, dst, dinv, bufA, E);
    }
    epilogue_kernel<<<(int)((featTot + B256 - 1) / B256), B256, 0, stream>>>(bufA, bufH, dinv, b2, featTot, 0);

    // ---- output layer + softmax (one warp per node) ----
    {
        int warpsPerBlock = 256 / 32;
        int blocks = (NN + warpsPerBlock - 1) / warpsPerBlock;
        out_softmax_kernel<<<blocks, 256, 0, stream>>>(bufA, Wout, bout, out, NN);
    }
}